// GCN_81810537054743
// MI455X (gfx1250) — compile-verified
//
#include <hip/hip_runtime.h>
#include <hip/hip_bf16.h>
#include <math.h>

#define NNODES 50000
#define NEDGES 800000
#define DIN    128
#define DH     256
#define DOUT   128
#define NEG_SLOPE 0.2f
#define L2EPS  1e-12f

typedef __attribute__((ext_vector_type(2))) float v2f;
typedef __attribute__((ext_vector_type(8))) float v8f;

// ---------------------------------------------------------------------------
// Dense GEMM + bias:  Y[nrows x NCOLS] = X[nrows x K] * W[K x NCOLS] + b
// One wave computes one 16x16 tile via V_WMMA_F32_16X16X4_F32.
// Block = 256 threads = 8 waves; the block shares a 16xK panel of X in LDS,
// staged with GLOBAL_LOAD_ASYNC_TO_LDS_B128 (ASYNCcnt path, no VGPR bounce).
// Requires nrows % 16 == 0; NCOLS covered as grid.y * (8 waves x 16).
// ---------------------------------------------------------------------------
template <int K, int NCOLS>
__launch_bounds__(256)
__global__ void gemm_bias_wmma(const float* __restrict__ X,
                               const float* __restrict__ W,
                               const float* __restrict__ bias,
                               float* __restrict__ Y)
{
    // Only __shared__ object in this kernel -> LDS base offset 0.
    __shared__ float As[16][K + 4];          // +4 pad -> conflict-free LDS reads

    const int m0   = blockIdx.x * 16;
    const int wave = threadIdx.x >> 5;
    const int lane = threadIdx.x & 31;
    const int half = lane >> 4;              // 0: lanes 0-15, 1: lanes 16-31
    const int lrow = lane & 15;
    const int n0   = (blockIdx.y * 8 + wave) * 16;

    // Async-stage the 16xK A-panel directly into LDS (16 bytes per lane-op).
    constexpr int F4 = K / 4;                // float4s per row
    for (int i = threadIdx.x; i < 16 * F4; i += 256) {
        const int r = i / F4;
        const int c = i - r * F4;
        const float*   gsrc = X + (size_t)(m0 + r) * K + c * 4;
        const unsigned ldst = (unsigned)(r * (K + 4) + c * 4) * 4u; // byte offset
        asm volatile("global_load_async_to_lds_b128 %0, %1, off"
                     :: "v"(ldst), "v"(gsrc)
                     : "memory");
    }
    asm volatile("s_wait_asynccnt 0x0" ::: "memory");
    __syncthreads();

    v8f acc = {};
    const int colIdx = n0 + lrow;

    #pragma unroll 4
    for (int k = 0; k < K; k += 4) {
        const int ks = k + half * 2;         // ISA f32 A/B layout: hi half-wave holds K+2,K+3
        v2f a, b;
        a.x = As[lrow][ks];
        a.y = As[lrow][ks + 1];
        b.x = W[(size_t)ks * NCOLS + colIdx];
        b.y = W[(size_t)(ks + 1) * NCOLS + colIdx];
        acc = __builtin_amdgcn_wmma_f32_16x16x4_f32(
                  false, a, false, b, (short)0, acc, false, false);
    }

    const float bv = bias[colIdx];
    #pragma unroll
    for (int v = 0; v < 8; ++v) {
        const int row = m0 + v + half * 8;   // C/D layout: VGPR v -> M=v (lo) / v+8 (hi)
        Y[(size_t)row * NCOLS + colIdx] = acc[v] + bv;
    }
}

// ---------------------------------------------------------------------------
// Utility kernels
// ---------------------------------------------------------------------------
__global__ void memset_f4(float4* __restrict__ p, int n4)
{
    const int i = blockIdx.x * blockDim.x + threadIdx.x;
    if (i < n4) p[i] = make_float4(0.f, 0.f, 0.f, 0.f);
}

__global__ void degree_count(const int* __restrict__ edges, float* __restrict__ cnt, int e)
{
    const int i = blockIdx.x * blockDim.x + threadIdx.x;
    if (i < e) atomicAdd(&cnt[edges[2 * i + 1]], 1.0f);
}

__global__ void degree_recip(float* __restrict__ inv, int n)
{
    const int i = blockIdx.x * blockDim.x + threadIdx.x;
    if (i < n) {
        const float c = inv[i];
        inv[i] = (c > 0.f) ? (1.0f / c) : 0.f;
    }
}

// Scatter-add: acc[dst] += sup[src]   (inv[dst] scaling deferred to next pass)
template <int F>
__launch_bounds__(256)
__global__ void scatter_add(const float* __restrict__ sup,
                            const int* __restrict__ edges,
                            float* __restrict__ acc)
{
    constexpr int F4 = F / 4;
    const int idx = blockIdx.x * blockDim.x + threadIdx.x;
    if (idx >= NEDGES * F4) return;
    const int e = idx / F4;                  // constant divisor
    const int c = idx - e * F4;
    const int src = edges[2 * e];
    const int dst = edges[2 * e + 1];
    const float4 v = reinterpret_cast<const float4*>(sup + (size_t)src * F)[c];
    float* d = acc + (size_t)dst * F + c * 4;
    atomicAdd(d + 0, v.x);
    atomicAdd(d + 1, v.y);
    atomicAdd(d + 2, v.z);
    atomicAdd(d + 3, v.w);
}

// h[i] = leaky(acc[i] * inv[row])   (in place)
template <int F>
__global__ void scale_leaky(float* __restrict__ h, const float* __restrict__ inv, int total)
{
    const int i = blockIdx.x * blockDim.x + threadIdx.x;
    if (i >= total) return;
    const float v = h[i] * inv[i / F];
    h[i] = (v > 0.f) ? v : NEG_SLOPE * v;
}

// Final: out[row] = (acc[row] * inv[row]) / max(||acc[row]*inv[row]||, eps)
// One wave per 128-float row; float4 per lane; wave32 shuffle reduction.
__launch_bounds__(256)
__global__ void final_norm(const float* __restrict__ acc,
                           const float* __restrict__ inv,
                           float* __restrict__ out)
{
    const int wave = threadIdx.x >> 5;
    const int lane = threadIdx.x & 31;
    const int row  = blockIdx.x * 8 + wave;
    if (row >= NNODES) return;

    const float s = inv[row];
    float4 v = reinterpret_cast<const float4*>(acc + (size_t)row * DOUT)[lane];
    v.x *= s; v.y *= s; v.z *= s; v.w *= s;

    float ss = v.x * v.x + v.y * v.y + v.z * v.z + v.w * v.w;
    #pragma unroll
    for (int o = 16; o > 0; o >>= 1) ss += __shfl_xor(ss, o, 32);

    const float sc = 1.0f / fmaxf(sqrtf(ss), L2EPS);
    v.x *= sc; v.y *= sc; v.z *= sc; v.w *= sc;
    reinterpret_cast<float4*>(out + (size_t)row * DOUT)[lane] = v;
}

// ---------------------------------------------------------------------------
// Launch
// ---------------------------------------------------------------------------
extern "C" void kernel_launch(void* const* d_in, const int* in_sizes, int n_in,
                              void* d_out, int out_size, void* d_ws, size_t ws_size,
                              hipStream_t stream)
{
    const float* x  = (const float*)d_in[0];
    const float* W1 = (const float*)d_in[1];
    const float* b1 = (const float*)d_in[2];
    const float* W2 = (const float*)d_in[3];
    const float* b2 = (const float*)d_in[4];
    const float* W3 = (const float*)d_in[5];
    const float* b3 = (const float*)d_in[6];
    const int* edges = (const int*)d_in[7];
    float* out = (float*)d_out;

    float* bufA = (float*)d_ws;                     // N x 256
    float* bufB = bufA + (size_t)NNODES * DH;       // N x 256
    float* inv  = bufB + (size_t)NNODES * DH;       // N

    const dim3 blk(256);
    auto cdiv = [](int a, int b) { return (a + b - 1) / b; };

    // --- degree -> 1/deg ---
    memset_f4<<<cdiv(NNODES / 4, 256), blk, 0, stream>>>((float4*)inv, NNODES / 4);
    degree_count<<<cdiv(NEDGES, 256), blk, 0, stream>>>(edges, inv, NEDGES);
    degree_recip<<<cdiv(NNODES, 256), blk, 0, stream>>>(inv, NNODES);

    // --- layer 1: support = x @ W1 + b1 ; h1 = leaky(inv * scatter(support)) ---
    gemm_bias_wmma<DIN, DH><<<dim3(NNODES / 16, DH / 128), blk, 0, stream>>>(x, W1, b1, bufA);
    memset_f4<<<cdiv(NNODES * DH / 4, 256), blk, 0, stream>>>((float4*)bufB, NNODES * DH / 4);
    scatter_add<DH><<<cdiv(NEDGES * (DH / 4), 256), blk, 0, stream>>>(bufA, edges, bufB);
    scale_leaky<DH><<<cdiv(NNODES * DH, 256), blk, 0, stream>>>(bufB, inv, NNODES * DH);

    // --- layer 2 ---
    gemm_bias_wmma<DH, DH><<<dim3(NNODES / 16, DH / 128), blk, 0, stream>>>(bufB, W2, b2, bufA);
    memset_f4<<<cdiv(NNODES * DH / 4, 256), blk, 0, stream>>>((float4*)bufB, NNODES * DH / 4);
    scatter_add<DH><<<cdiv(NEDGES * (DH / 4), 256), blk, 0, stream>>>(bufA, edges, bufB);
    scale_leaky<DH><<<cdiv(NNODES * DH, 256), blk, 0, stream>>>(bufB, inv, NNODES * DH);

    // --- layer 3 + L2 normalize ---
    gemm_bias_wmma<DH, DOUT><<<dim3(NNODES / 16, DOUT / 128), blk, 0, stream>>>(bufB, W3, b3, bufA);
    memset_f4<<<cdiv(NNODES * DOUT / 4, 256), blk, 0, stream>>>((float4*)bufB, NNODES * DOUT / 4);
    scatter_add<DOUT><<<cdiv(NEDGES * (DOUT / 4), 256), blk, 0, stream>>>(bufA, edges, bufB);
    final_norm<<<cdiv(NNODES, 8), blk, 0, stream>>>(bufB, inv, out);
}